// LoRARouter_60120952209600
// MI455X (gfx1250) — compile-verified
//
#include <hip/hip_runtime.h>
#include <cstdint>

#define BB 8
#define SS 2048
#define HH 4096

typedef float v2f __attribute__((ext_vector_type(2)));
typedef float v8f __attribute__((ext_vector_type(8)));

constexpr int RP = 20;     // tile row pitch (dwords): 20*L mod 64 steps by 4 -> conflict-free b64 reads
constexpr int NBUF = 4;    // quad-buffer: prefetch distance 3 tiles (3 KB in flight per wave)

// ---------------------------------------------------------------------------
// Prep: per-batch mask count + last-token dot with W2 (= W[H..2H))
// ---------------------------------------------------------------------------
__global__ __launch_bounds__(256) void prep_kernel(
    const float* __restrict__ hs, const int* __restrict__ mask,
    const float* __restrict__ W, float* __restrict__ dotlast, int* __restrict__ cnt)
{
  const int b = blockIdx.x;
  const int t = threadIdx.x;
  __shared__ int   scnt[256];
  __shared__ float sdot[256];
  __shared__ int   idx_sh;

  int c = 0;
  for (int s = t; s < SS; s += 256) c += mask[b * SS + s];
  scnt[t] = c;
  __syncthreads();
  for (int off = 128; off > 0; off >>= 1) {
    if (t < off) scnt[t] += scnt[t + off];
    __syncthreads();
  }
  if (t == 0) {
    int n = scnt[0];
    cnt[b] = n;
    int idx = n - 1;
    if (idx < 0) idx += SS;   // JAX negative-index wrap
    idx_sh = idx;
  }
  __syncthreads();

  const float* row = hs + ((size_t)b * SS + idx_sh) * HH;
  const float* w2  = W + HH;
  float d = 0.f;
  for (int h = t; h < HH; h += 256) d += row[h] * w2[h];
  sdot[t] = d;
  __syncthreads();
  for (int off = 128; off > 0; off >>= 1) {
    if (t < off) sdot[t] += sdot[t + off];
    __syncthreads();
  }
  if (t == 0) dotlast[b] = sdot[0];
}

// ---------------------------------------------------------------------------
// Main streaming pass: acc[b] += sum_s mask[b,s] * (hidden[b,s,:] . W1)
// WMMA f32 16x16x4, A staged via async global->LDS, 4-deep pipeline with a
// branch-free steady-state loop and a peeled 3-tile drain epilogue.
// ---------------------------------------------------------------------------
__global__ __launch_bounds__(256) void pool_wmma_kernel(
    const float* __restrict__ hs, const int* __restrict__ mask,
    const float* __restrict__ W, float* __restrict__ acc)
{
  __shared__ __align__(16) float wlds[HH];                    // 16 KB copy of W1
  __shared__ __align__(16) float tiles[8][NBUF][16 * RP];     // per-wave quad-buffered 16x16 tiles

  const int t = threadIdx.x;
  for (int i = t; i < HH; i += 256) wlds[i] = W[i];
  __syncthreads();

  const int batch = blockIdx.x >> 7;   // 128 strips per batch
  const int strip = blockIdx.x & 127;
  const int s0    = strip << 4;        // 16 s-rows per strip
  const int wv    = t >> 5;
  const int lane  = t & 31;
  const int grp   = lane >> 4;         // 0: lanes 0-15 (K=0,1 / M=r), 1: lanes 16-31 (K=2,3 / M=r+8)
  const int arow  = lane & 15;

  const float* base = hs + ((size_t)batch * SS + s0) * HH;

  auto issue = [&](int g) {
    const int bsel = g & (NBUF - 1);
    const int hb = (wv << 4) + (g << 7);                       // this wave's 16-col group
    unsigned lbase = (unsigned)(uintptr_t)&tiles[wv][bsel][0];
#pragma unroll
    for (int i = 0; i < 2; ++i) {
      int q   = lane + (i << 5);       // 0..63 -> 16 rows x 4 quads
      int row = q >> 2;
      int qc  = q & 3;
      unsigned long long ga =
          (unsigned long long)(uintptr_t)(base + (size_t)row * HH + hb + (qc << 2));
      unsigned la = lbase + (unsigned)(row * RP + (qc << 2)) * 4u;
      asm volatile("global_load_async_to_lds_b128 %0, %1, off"
                   :: "v"(la), "v"(ga) : "memory");
    }
  };

  v8f c = {0.f, 0.f, 0.f, 0.f, 0.f, 0.f, 0.f, 0.f};

  auto compute = [&](int g) {
    const float* tp = &tiles[wv][g & (NBUF - 1)][0];
    const int hb = (wv << 4) + (g << 7);
#pragma unroll
    for (int cc = 0; cc < 16; cc += 4) {
      // A: 16x4 f32 chunk; lane L reads tile[row=L%16][cc + 2*(L/16)] (2 floats)
      v2f a = *(const v2f*)(tp + arow * RP + cc + 2 * grp);
      // B: W1[hb+cc+k] replicated across all 16 N columns (same per-lane layout)
      v2f b = *(const v2f*)(wlds + hb + cc + 2 * grp);
      c = __builtin_amdgcn_wmma_f32_16x16x4_f32(
          /*neg_a=*/false, a, /*neg_b=*/false, b,
          /*c_mod=*/(short)0, c, /*reuse_a=*/false, /*reuse_b=*/false);
    }
  };

  // Prologue: fill the pipeline 3 tiles deep (6 outstanding async loads/wave).
  issue(0);
  issue(1);
  issue(2);

  // Steady state: branch-free. Tile g consumed while g+1..g+3 are in flight.
  for (int g = 0; g < 29; ++g) {
    issue(g + 3);
    asm volatile("s_wait_asynccnt 0x6" ::: "memory");
    compute(g);
  }

  // Drain epilogue (peeled, straight-line).
  asm volatile("s_wait_asynccnt 0x4" ::: "memory");
  compute(29);
  asm volatile("s_wait_asynccnt 0x2" ::: "memory");
  compute(30);
  asm volatile("s_wait_asynccnt 0x0" ::: "memory");
  compute(31);

  // C layout: VGPR r, lanes 0-15 -> row m=r ; lanes 16-31 -> row m=r+8 (cols identical).
  // Weight each s-row by its mask, then total = lane0 (m=0..7) + lane16 (m=8..15).
  const int* mrow = mask + batch * SS + s0;
  float sum = 0.f;
#pragma unroll
  for (int r = 0; r < 8; ++r)
    sum += c[r] * (float)mrow[r + (grp << 3)];

  float tot = __shfl(sum, 0, 32) + __shfl(sum, 16, 32);
  if (lane == 0) atomicAdd(&acc[batch], tot);
}

// ---------------------------------------------------------------------------
// Finalize: sigmoid(acc/clip(cnt) + dotlast + bias)
// ---------------------------------------------------------------------------
__global__ void zero_acc(float* acc) {
  if (threadIdx.x < BB) acc[threadIdx.x] = 0.f;
}

__global__ void finalize_kernel(const float* __restrict__ acc, const float* __restrict__ dl,
                                const int* __restrict__ cnt, const float* __restrict__ bias,
                                float* __restrict__ out)
{
  int b = threadIdx.x;
  if (b < BB) {
    float sm = (float)cnt[b];
    if (sm < 1e-9f) sm = 1e-9f;
    float x = acc[b] / sm + dl[b] + bias[0];
    out[b] = 1.f / (1.f + expf(-x));
  }
}

extern "C" void kernel_launch(void* const* d_in, const int* in_sizes, int n_in,
                              void* d_out, int out_size, void* d_ws, size_t ws_size,
                              hipStream_t stream) {
  (void)in_sizes; (void)n_in; (void)out_size; (void)ws_size;
  const float* hs   = (const float*)d_in[0];
  const int*   mask = (const int*)d_in[1];
  const float* W    = (const float*)d_in[2];
  const float* bias = (const float*)d_in[3];
  float* out = (float*)d_out;

  float* acc = (float*)d_ws;        // [8]
  float* dl  = acc + BB;            // [8]
  int*   cnt = (int*)(acc + 2 * BB);// [8]

  zero_acc<<<1, 32, 0, stream>>>(acc);
  prep_kernel<<<BB, 256, 0, stream>>>(hs, mask, W, dl, cnt);
  pool_wmma_kernel<<<BB * (SS / 16), 256, 0, stream>>>(hs, mask, W, acc);
  finalize_kernel<<<1, 32, 0, stream>>>(acc, dl, cnt, bias, out);
}